// KnnSearch_71451075936753
// MI455X (gfx1250) — compile-verified
//
#include <hip/hip_runtime.h>

typedef float v2f __attribute__((ext_vector_type(2)));
typedef float v4f __attribute__((ext_vector_type(4)));
typedef float v8f __attribute__((ext_vector_type(8)));

#define BATCH 4
#define NSUP  16384
#define NQRY  4096
#define KNN   16
#define CHUNK 2048
#define WAVES 8
#define BLOCK (WAVES * 32)
#define QPW   16               // queries per wave (one WMMA column block)
#define QPB   (WAVES * QPW)    // 128 queries per block

__global__ __launch_bounds__(BLOCK)
void knn_wmma_kernel(const float* __restrict__ sup,
                     const float* __restrict__ qry,
                     int* __restrict__ out)
{
    // Support chunk staged in LDS: float4 {x,y,z,0} per point + squared norms.
    __shared__ float sPts[CHUNK * 4];
    __shared__ float sS2[CHUNK];

    const int tid  = threadIdx.x;
    const int lane = tid & 31;
    const int wave = tid >> 5;
    const int lo   = lane & 15;   // query column within the 16x16 tile
    const int hi   = lane >> 4;   // which 8-row half of the tile this lane owns

    const int b      = blockIdx.x >> 5;            // 32 blocks per batch
    const int qBlock = (blockIdx.x & 31) * QPB;
    const int qn     = qBlock + wave * QPW + lo;   // this lane's query index

    // ---- Load query, build B operand (4x16, K-major per documented layout) ----
    const float* qp = qry + ((size_t)b * NQRY + qn) * 3;
    const float qx = qp[0], qy = qp[1], qz = qp[2];
    v2f Bop;
    Bop.x = hi ? qz : qx;      // VGPR0: K=0 (lanes 0-15) / K=2 (lanes 16-31)
    Bop.y = hi ? 0.0f : qy;    // VGPR1: K=1 (lanes 0-15) / K=3 (lanes 16-31)

    // Per-lane top-K over dv = s2 - 2*q.s  (== dist2 - q2; q2 is constant per
    // lane AND identical on the shuffle-partner lane, so ordering is the same
    // and only indices are output). Slot 0 always holds the current max.
    float bd[KNN];
    int   bi[KNN];
#pragma unroll
    for (int j = 0; j < KNN; ++j) { bd[j] = 3.4e38f; bi[j] = 0; }

    const float* sbase = sup + (size_t)b * NSUP * 3;
    const int aoff  = (lo << 2) + (hi << 1);   // float offset of A fragment
    const int s2off = hi << 3;                 // float offset of s2 quad pair

    for (int cb = 0; cb < NSUP; cb += CHUNK) {
        __syncthreads();
        // ---- Stage CHUNK support points into LDS (coalesced), precompute s2 ----
#pragma unroll
        for (int j = 0; j < CHUNK / BLOCK; ++j) {
            const int i = tid + j * BLOCK;
            const float* sp = sbase + (size_t)(cb + i) * 3;
            const float x = sp[0], y = sp[1], z = sp[2];
            v4f p; p.x = x; p.y = y; p.z = z; p.w = 0.0f;
            *(v4f*)&sPts[i << 2] = p;          // single ds_store_b128
            sS2[i] = x * x + y * y + z * z;
        }
        // Hint-prefetch the next chunk while we compute on this one.
        if (cb + CHUNK < NSUP)
            __builtin_prefetch(sbase + (size_t)(cb + CHUNK + tid * 8) * 3, 0, 0);
        __syncthreads();

        // ---- Software-pipelined sweep of 16-support tiles ----
        // Preload tile 0 operands.
        v2f Anext = *(const v2f*)&sPts[aoff];
        v4f s2na  = *(const v4f*)&sS2[s2off];
        v4f s2nb  = *(const v4f*)&sS2[s2off + 4];

        for (int t = 0; t < CHUNK; t += 16) {
            const v2f Acur = Anext;
            const v4f s2a  = s2na;
            const v4f s2b  = s2nb;
            // Issue next tile's LDS loads early so candidate processing
            // hides their latency (DScnt is in-order -> partial waits).
            const int tn = t + 16;
            if (tn < CHUNK) {
                Anext = *(const v2f*)&sPts[(tn << 2) + aoff];
                s2na  = *(const v4f*)&sS2[tn + s2off];
                s2nb  = *(const v4f*)&sS2[tn + s2off + 4];
            }

            v8f acc = {};
            acc = __builtin_amdgcn_wmma_f32_16x16x4_f32(
                false, Acur, false, Bop, (short)0, acc, false, false);

            const int rbase = t + (hi << 3);   // this lane's 8 support rows
            float s2v[8];
            s2v[0] = s2a.x; s2v[1] = s2a.y; s2v[2] = s2a.z; s2v[3] = s2a.w;
            s2v[4] = s2b.x; s2v[5] = s2b.y; s2v[6] = s2b.z; s2v[7] = s2b.w;

#pragma unroll
            for (int v = 0; v < 8; ++v) {
                const float dv = __builtin_fmaf(-2.0f, acc[v], s2v[v]);
                if (dv < bd[0]) {              // rare: guarded insert
                    bd[0] = dv; bi[0] = cb + rbase + v;
#pragma unroll
                    for (int j = 1; j < KNN; ++j) {  // restore max at slot 0
                        const bool  sw = bd[j] > bd[0];
                        const float td = bd[0]; const int ti = bi[0];
                        bd[0] = sw ? bd[j] : bd[0]; bi[0] = sw ? bi[j] : bi[0];
                        bd[j] = sw ? td    : bd[j]; bi[j] = sw ? ti    : bi[j];
                    }
                }
            }
        }
    }

    // ---- Merge the two half-lists for each query: lane l+16 -> lane l ----
#pragma unroll
    for (int j = 0; j < KNN; ++j) {
        const float pdist = __shfl_down(bd[j], 16, 32);
        const int   pidx  = __shfl_down(bi[j], 16, 32);
        if (hi == 0 && pdist < bd[0]) {
            bd[0] = pdist; bi[0] = pidx;
#pragma unroll
            for (int m = 1; m < KNN; ++m) {
                const bool  sw = bd[m] > bd[0];
                const float td = bd[0]; const int ti = bi[0];
                bd[0] = sw ? bd[m] : bd[0]; bi[0] = sw ? bi[m] : bi[0];
                bd[m] = sw ? td    : bd[m]; bi[m] = sw ? ti    : bi[m];
            }
        }
    }

    // ---- Sort ascending (unrolled insertion network, constant indices) ----
#pragma unroll
    for (int i = 1; i < KNN; ++i) {
#pragma unroll
        for (int j = KNN - 1; j > 0; --j) {
            if (j > i) continue;
            const bool  sw = bd[j] < bd[j - 1];
            const float td = bd[j - 1]; const int ti = bi[j - 1];
            bd[j - 1] = sw ? bd[j] : bd[j - 1]; bi[j - 1] = sw ? bi[j] : bi[j - 1];
            bd[j]     = sw ? td    : bd[j];     bi[j]     = sw ? ti    : bi[j];
        }
    }

    // ---- Lanes 0-15 own the merged, sorted result: write indices ----
    if (hi == 0) {
        int* op = out + ((size_t)b * NQRY + qn) * KNN;
#pragma unroll
        for (int j = 0; j < KNN; ++j) op[j] = bi[j];
    }
}

extern "C" void kernel_launch(void* const* d_in, const int* in_sizes, int n_in,
                              void* d_out, int out_size, void* d_ws, size_t ws_size,
                              hipStream_t stream) {
    (void)in_sizes; (void)n_in; (void)out_size; (void)d_ws; (void)ws_size;
    const float* sup = (const float*)d_in[0];   // [4, 16384, 3] f32
    const float* qry = (const float*)d_in[1];   // [4, 4096, 3]  f32
    int* out = (int*)d_out;                     // [4, 4096, 16] i32
    knn_wmma_kernel<<<dim3(BATCH * (NQRY / QPB)), dim3(BLOCK), 0, stream>>>(sup, qry, out);
}